// TorchRoughSABR_FMM_22789096473346
// MI455X (gfx1250) — compile-verified
//
#include <hip/hip_runtime.h>
#include <math.h>
#include <stdint.h>

typedef __attribute__((ext_vector_type(8))) float v8f;
typedef __attribute__((ext_vector_type(2))) float v2f;

constexpr int P = 4096, S = 512, N = 40, NP = 48;

// ---- CDNA5 async memory->LDS copies (ASYNCcnt-tracked, no VGPR round-trip) ----
__device__ __forceinline__ uint32_t lds_off(const void* p) {
  return (uint32_t)(uintptr_t)p;  // low 32 bits of a flat LDS pointer = LDS byte offset
}
__device__ __forceinline__ void async_ld_b128(uint32_t lds, const void* g) {
  asm volatile("global_load_async_to_lds_b128 %0, %1, off" :: "v"(lds), "v"(g) : "memory");
}
__device__ __forceinline__ void async_ld_b32(uint32_t lds, const void* g) {
  asm volatile("global_load_async_to_lds_b32 %0, %1, off" :: "v"(lds), "v"(g) : "memory");
}
__device__ __forceinline__ void wait_async0() {
  asm volatile("s_wait_asynccnt 0x0" ::: "memory");
}
__device__ __forceinline__ void wait_async2() {
  asm volatile("s_wait_asynccnt 0x2" ::: "memory");
}

// ---------- Kernel 1: pair-interleaved kernel^T + var_comp ----------
// kTP[((j>>1)*S + i)*2 + (j&1)] = kernel[i][j]   (so B tiles async-copy as contiguous pairs)
__global__ void k_precompute(const float* __restrict__ tg,
                             const float* __restrict__ nus,
                             const float* __restrict__ Hp,
                             float* __restrict__ kTP,
                             float* __restrict__ var_comp) {
  int idx = blockIdx.x * blockDim.x + threadIdx.x;
  float H  = Hp[0];
  float dt = tg[1] - tg[0];
  float gc = expf(lgammaf(H + 0.5f));
  if (idx < S * S) {
    int j = idx / S, i = idx % S;
    float v = 0.0f;
    if (i >= j) v = powf((float)(i + 1 - j) * dt, H - 0.5f) / gc;
    kTP[(((size_t)(j >> 1)) * S + i) * 2 + (j & 1)] = v;
  }
  if (idx < S) {
    float nu = nus[0];
    float t  = (float)(idx + 1) * dt;
    var_comp[idx] = 0.5f * nu * nu * powf(t, 2.0f * H) / (2.0f * H * gc * gc);
  }
}

// ---------- Kernel 2: dW_v[p][j] ----------
__global__ void k_dwv(const float* __restrict__ z,
                      const float* __restrict__ tg,
                      const float* __restrict__ rhos,
                      float* __restrict__ dWv) {
  int idx = blockIdx.x * blockDim.x + threadIdx.x;  // p*S + j
  if (idx >= P * S) return;
  float dt  = tg[1] - tg[0];
  float sdt = sqrtf(dt);
  float rho = rhos[0];
  float c2  = sqrtf(1.0f - rho * rho);
  const float4 z4 = *reinterpret_cast<const float4*>(z + (size_t)idx * 4);
  dWv[idx] = (rho * z4.x + c2 * z4.w) * sdt;
}

// ---------- Kernel 3: fBm = dWv @ kernel^T (f32 WMMA, async double-buffered staging) ----------
__global__ __launch_bounds__(128) void k_fbm_gemm(
    const float* __restrict__ dWv,      // [P][S]
    const float* __restrict__ kTP,      // pair-interleaved [S/2][S][2]
    const float* __restrict__ var_comp, // [S]
    const float* __restrict__ nus,
    float* __restrict__ uv)             // [P][S]
{
  __shared__ float As[2][64 * 18];  // 64 rows(p) x 16 cols(j); row stride 18 floats (72B)
  __shared__ v2f   Bs2[2][8 * 64];  // 8 pair-rows x 64 cols of {B[2k][i],B[2k+1][i]}
  const int tid  = threadIdx.x;
  const int lane = tid & 31, wave = tid >> 5;
  const int half = lane >> 4, lm = lane & 15;
  const int p0 = blockIdx.x * 64;
  const int i0 = blockIdx.y * 64;

  v8f acc[4] = {};
  const int ktEnd = (S < i0 + 64) ? S : (i0 + 64);  // kernel^T[j][i]==0 for j>i

  // async-stage one 16-deep K tile into buffer b
  auto stage = [&](int j0, int b) {
    uint32_t aB = lds_off(&As[b][0]);
    uint32_t bB = lds_off(&Bs2[b][0]);
    #pragma unroll
    for (int q = tid; q < 256; q += 128) {          // A: 64x16, 16B chunks within a row
      int r = q >> 2, c4 = q & 3;
      async_ld_b128(aB + r * 72 + c4 * 16, dWv + (size_t)(p0 + r) * S + j0 + c4 * 4);
    }
    #pragma unroll
    for (int q = tid; q < 256; q += 128) {          // B: 8 pair-rows x 512B, contiguous
      int j2 = q >> 5, c = q & 31;
      async_ld_b128(bB + j2 * 512 + c * 16,
                    kTP + ((size_t)(j0 / 2 + j2) * S + i0) * 2 + c * 4);
    }
  };

  stage(0, 0);
  wait_async0();
  __syncthreads();

  int cur = 0;
  for (int j0 = 0; j0 < ktEnd; j0 += 16) {
    if (j0 + 16 < ktEnd) stage(j0 + 16, cur ^ 1);   // prefetch next tile while computing
    #pragma unroll
    for (int c = 0; c < 4; ++c) {
      v2f a;
      const float* ap = &As[cur][(wave * 16 + lm) * 18 + c * 4 + 2 * half];
      a.x = ap[0]; a.y = ap[1];
      #pragma unroll
      for (int nt = 0; nt < 4; ++nt) {
        v2f b = Bs2[cur][(c * 2 + half) * 64 + nt * 16 + lm];  // one ds_load_b64, paired
        acc[nt] = __builtin_amdgcn_wmma_f32_16x16x4_f32(
            false, a, false, b, (short)0, acc[nt], false, false);
      }
    }
    wait_async0();       // next tile's async writes landed
    __syncthreads();     // everyone done reading current tile
    cur ^= 1;
  }

  const float nu = nus[0];
  #pragma unroll
  for (int nt = 0; nt < 4; ++nt) {
    int i = i0 + nt * 16 + lm;
    float vc = var_comp[i];
    #pragma unroll
    for (int e = 0; e < 8; ++e) {
      int p = p0 + wave * 16 + e + 8 * half;       // C-layout: row = e + 8*half, col = lane%16
      uv[(size_t)p * S + i] = expf(nu * acc[nt][e] - vc);
    }
  }
}

// ---------- Kernel 4: sequential scan; F in WMMA C-layout; x@Lam^T via f32 WMMA ----------
__global__ __launch_bounds__(32) void k_scan(
    const float* __restrict__ z,        // [P][S][4]
    const float* __restrict__ tg,
    const float* __restrict__ F0,
    const float* __restrict__ alphas,
    const float* __restrict__ taus,
    const float* __restrict__ loadings, // [N][3]
    const float* __restrict__ uv,       // [P][S]
    float* __restrict__ out)            // [P][S+1][N]
{
  __shared__ float  Bsh[NP * NP];       // Bsh[k*48+n] = Lam[n][k] (nonzero for k>n), zero-padded
  __shared__ float  xS[16 * 50];        // 16 paths x 48 cols, stride 50 (conflict-free, even)
  __shared__ float4 zS[2][16];          // raw z[p][s][0..3], async double buffer
  __shared__ float  uS[2][16];          // uv[p][s], async double buffer
  const int lane = threadIdx.x & 31;
  const int half = lane >> 4, lm = lane & 15;
  const int pBase = blockIdx.x * 16;
  const float dt  = tg[1] - tg[0];
  const float sdt = sqrtf(dt);

  for (int idx = threadIdx.x; idx < NP * NP; idx += 32) {
    int k = idx / NP, n = idx % NP;
    float v = 0.0f;
    if (k < N && n < N && k > n)
      v = loadings[n*3+0]*loadings[k*3+0] + loadings[n*3+1]*loadings[k*3+1]
        + loadings[n*3+2]*loadings[k*3+2];
    Bsh[idx] = v;
  }
  __syncthreads();

  // per-(lane, tile) column constants; sdt folded into loadings; padded cols (n>=40) -> zeros
  float alC[3], ctaC[3], ctC[3], l0C[3], l1C[3], l2C[3];
  v8f F[3];
  #pragma unroll
  for (int t = 0; t < 3; ++t) {
    int n = t * 16 + lm;
    float a = 0.f, tn = 0.f, L0 = 0.f, L1 = 0.f, L2 = 0.f, f = 0.f;
    if (n < N) {
      a = alphas[n]; tn = taus[n];
      L0 = loadings[n*3+0] * sdt; L1 = loadings[n*3+1] * sdt; L2 = loadings[n*3+2] * sdt;
      f = F0[n];
    }
    alC[t] = a; ctC[t] = tn; ctaC[t] = tn * a;
    l0C[t] = L0; l1C[t] = L1; l2C[t] = L2;
    #pragma unroll
    for (int e = 0; e < 8; ++e) F[t][e] = f;
  }

  // Lam^T fragments (constant across all 512 steps) -> registers; triangular pruning: 18 live
  v2f bf[10][3];
  #pragma unroll
  for (int c = 0; c < 10; ++c) {
    #pragma unroll
    for (int t = 0; t < 3; ++t) {
      if (4*c + 3 > 16*t) {
        bf[c][t].x = Bsh[(4*c + 2*half)     * NP + 16*t + lm];
        bf[c][t].y = Bsh[(4*c + 2*half + 1) * NP + 16*t + lm];
      }
    }
  }

  // initial output row (time index 0) = F0
  #pragma unroll
  for (int e = 0; e < 8; ++e) {
    int p = pBase + e + 8*half;
    #pragma unroll
    for (int t = 0; t < 3; ++t) {
      int n = t*16 + lm;
      if (t < 2 || lm < 8)
        __builtin_nontemporal_store(F[t][e], &out[(size_t)p * (S+1) * N + n]);
    }
  }

  // async prefetch of step 0 (z row is copied verbatim: 16B per path)
  if (lane < 16) {
    int p = pBase + lane;
    async_ld_b128(lds_off(&zS[0][lane]), z + ((size_t)p * S + 0) * 4);
    async_ld_b32 (lds_off(&uS[0][lane]), uv + (size_t)p * S + 0);
  }

  for (int s = 0; s < S; ++s) {
    // issue next step's async copies (kept in flight behind this step's WMMA work)
    {
      int sn = (s + 1 < S) ? s + 1 : s;
      int nb = (s + 1) & 1;
      if (lane < 16) {
        int p = pBase + lane;
        async_ld_b128(lds_off(&zS[nb][lane]), z + ((size_t)p * S + sn) * 4);
        async_ld_b32 (lds_off(&uS[nb][lane]), uv + (size_t)p * S + sn);
      }
    }
    // x = tau*alpha / (1 + tau*F), staged to LDS for C-layout -> A-layout transpose
    #pragma unroll
    for (int t = 0; t < 3; ++t) {
      #pragma unroll
      for (int e = 0; e < 8; ++e) {
        float xv = __fdividef(ctaC[t], 1.0f + ctC[t] * F[t][e]);
        xS[(e + 8*half) * 50 + 16*t + lm] = xv;       // same-wave DS ops are in-order
      }
    }
    // split even/odd-chunk accumulators: halves the serial WMMA chain depth per tile
    v8f accA[3] = {}, accB[3] = {};
    #pragma unroll
    for (int c = 0; c < 10; ++c) {
      v2f a;
      const float* ap = &xS[lm * 50 + 4*c + 2*half];  // A-layout: m=lane%16, k split by half
      a.x = ap[0]; a.y = ap[1];
      #pragma unroll
      for (int t = 0; t < 3; ++t) {
        if (4*c + 3 > 16*t) {
          if (c & 1)
            accB[t] = __builtin_amdgcn_wmma_f32_16x16x4_f32(
                false, a, false, bf[c][t], (short)0, accB[t], false, false);
          else
            accA[t] = __builtin_amdgcn_wmma_f32_16x16x4_f32(
                false, a, false, bf[c][t], (short)0, accA[t], false, false);
        }
      }
    }
    wait_async2();   // retire this step's 2 copies; next step's 2 stay in flight
    // F_new = F - alpha*(x@Lam^T)*v2 + dW_r*uv*alpha ; write out[p][s+1][n]
    const int cb = s & 1;
    #pragma unroll
    for (int e = 0; e < 8; ++e) {
      float4 q = zS[cb][e + 8*half];                  // LDS broadcast read (raw z)
      float u  = uS[cb][e + 8*half];
      float v2 = u * u * dt;
      int p = pBase + e + 8*half;
      #pragma unroll
      for (int t = 0; t < 3; ++t) {
        float y = accA[t][e] + accB[t][e];
        float dwr = q.x*l0C[t] + q.y*l1C[t] + q.z*l2C[t];  // sdt pre-folded into lC
        float fn = F[t][e] + alC[t] * (u * dwr - y * v2);
        F[t][e] = fn;
        int n = t*16 + lm;
        if (t < 2 || lm < 8)
          __builtin_nontemporal_store(fn, &out[((size_t)p * (S+1) + (s+1)) * N + n]);
      }
    }
  }
}

extern "C" void kernel_launch(void* const* d_in, const int* in_sizes, int n_in,
                              void* d_out, int out_size, void* d_ws, size_t ws_size,
                              hipStream_t stream) {
  const float* z   = (const float*)d_in[0];
  const float* tg  = (const float*)d_in[1];
  const float* F0  = (const float*)d_in[2];
  const float* al  = (const float*)d_in[3];
  const float* rho = (const float*)d_in[4];
  const float* nu  = (const float*)d_in[5];
  const float* tau = (const float*)d_in[6];
  const float* ld  = (const float*)d_in[7];
  const float* H   = (const float*)d_in[8];
  float* out = (float*)d_out;

  // workspace: dWv (8MB) | kTP (1MB) | uv (8MB) | var_comp (2KB)  ~= 17.8 MB
  char* ws = (char*)d_ws;
  float* dWv   = (float*)(ws);
  float* kTP   = (float*)(ws + (size_t)P * S * 4);
  float* uvArr = (float*)(ws + (size_t)P * S * 4 + (size_t)S * S * 4);
  float* varc  = (float*)(ws + (size_t)P * S * 4 + (size_t)S * S * 4 + (size_t)P * S * 4);

  k_precompute<<<(S * S + 255) / 256, 256, 0, stream>>>(tg, nu, H, kTP, varc);
  k_dwv<<<(P * S + 255) / 256, 256, 0, stream>>>(z, tg, rho, dWv);
  dim3 g3(P / 64, S / 64);
  k_fbm_gemm<<<g3, 128, 0, stream>>>(dWv, kTP, varc, nu, uvArr);
  k_scan<<<P / 16, 32, 0, stream>>>(z, tg, F0, al, tau, ld, uvArr, out);
}